// HierDDpm_25692494364846
// MI455X (gfx1250) — compile-verified
//
#include <hip/hip_runtime.h>

#define S_LEN 8192
#define B_N   8
#define IN_D  128
#define D_D   64
#define NW    4          // waves per block
#define XS    132        // x-buffer row stride (pad vs 64 banks)
#define TS    68         // t-buffer row stride
#define WAVE_F (16*XS + 16*TS + 32)

typedef float v2f __attribute__((ext_vector_type(2)));
typedef float v8f __attribute__((ext_vector_type(8)));

__device__ __forceinline__ v8f wmma4(v2f a, v2f b, v8f c) {
  // D = A(16x4,f32) * B(4x16,f32) + C(16x16,f32)
  return __builtin_amdgcn_wmma_f32_16x16x4_f32(false, a, false, b, (short)0, c, false, false);
}

__device__ __forceinline__ void lds_fence() {
  asm volatile("s_wait_dscnt 0" ::: "memory");
}

// cos(2*pi*x): v_fract (exact range reduction to [0,1) revolutions) + v_cos
__device__ __forceinline__ float cos2pi(float x) {
  return __builtin_amdgcn_cosf(__builtin_amdgcn_fractf(x));
}

// One layer for TWO positions packed into the 16 WMMA rows:
// rows 0..7 = batch of s0, rows 8..15 = batch of s1.
template<int KIN, bool HAS_WR, bool LAST>
__device__ __forceinline__ void layer_step(
    float* xb, float* tb, float* st,
    const float* __restrict__ M, const float* __restrict__ P,
    const float* __restrict__ g, const float* __restrict__ bb,
    const float* __restrict__ Wr,
    int s0, int lane, float* __restrict__ outg)
{
  const int r  = lane & 15;   // A-operand row
  const int hi = lane >> 4;   // K-pair select (A/B) ; row-half select (C/D)
  const int n  = lane & 15;   // column within 16-wide tile

  v8f zero = {0.f,0.f,0.f,0.f,0.f,0.f,0.f,0.f};
  v8f acc[4], accR[4];
  #pragma unroll
  for (int t = 0; t < 4; ++t) { acc[t] = zero; accR[t] = zero; }

  // ---- dense: xt = X @ M^T (both positions share B), layer1 also X @ Wr^T ----
  for (int k0 = 0; k0 < KIN; k0 += 4) {
    int ka = k0 + 2*hi;
    v2f a = *(const v2f*)(xb + r*XS + ka);          // all 16 rows valid
    #pragma unroll
    for (int t = 0; t < 4; ++t) {
      int nt = t*16 + n;
      v2f bm = *(const v2f*)(M + nt*KIN + ka);
      acc[t] = wmma4(a, bm, acc[t]);
      if (HAS_WR) {
        v2f br = *(const v2f*)(Wr + nt*KIN + ka);
        accR[t] = wmma4(a, br, accR[t]);
      }
    }
  }

  // ---- stage xt (all 32 lanes, rows v+8*hi) ----
  #pragma unroll
  for (int t = 0; t < 4; ++t)
    #pragma unroll
    for (int v = 0; v < 8; ++v)
      tb[(v + 8*hi)*TS + t*16 + n] = acc[t][v];
  lds_fence();

  // ---- LayerNorm stats: one row per lane 0..15 (biased var, eps in sqrt) ----
  if (lane < 16) {
    float sm = 0.f, sq = 0.f;
    for (int k = 0; k < 64; ++k) { float x = tb[lane*TS + k]; sm += x; sq += x*x; }
    float mu  = sm * (1.f/64.f);
    float var = sq * (1.f/64.f) - mu*mu;
    st[lane]      = mu;
    st[16 + lane] = __builtin_amdgcn_rsqf(var + 1e-5f);
  }
  lds_fence();

  // ---- normalize in place, apply gamma/beta ----
  for (int idx = lane; idx < 16*64; idx += 32) {
    int rr = idx >> 6, k = idx & 63;
    float x = tb[rr*TS + k];
    tb[rr*TS + k] = (x - st[rr]) * st[16 + rr] * g[k] + bb[k];
  }
  lds_fence();

  // ---- modulated GEMMs: y_p = LN(xt) @ W_{s_p}^T + residual,
  //      W_s[i][j] = P[i][j] * cos(2*pi*s / (64*i + j + 2)).
  //      A registers shared across both positions; rcp(1/p) shared too. ----
  v8f accO0[4], accO1[4];
  if (HAS_WR) {
    #pragma unroll
    for (int t = 0; t < 4; ++t) { accO0[t] = accR[t]; accO1[t] = accR[t]; }
  } else {
    #pragma unroll
    for (int t = 0; t < 4; ++t) {
      #pragma unroll
      for (int v = 0; v < 8; ++v) {
        float rv = xb[(v + 8*hi)*XS + t*16 + n];   // residual = layer input
        accO0[t][v] = rv;
        accO1[t][v] = rv;
      }
    }
  }

  const float s0f = (float)s0;
  const float s1f = (float)(s0 + 1);
  for (int k0 = 0; k0 < 64; k0 += 4) {
    int ka = k0 + 2*hi;
    v2f a = *(const v2f*)(tb + r*TS + ka);
    #pragma unroll
    for (int t = 0; t < 4; ++t) {
      int nt = t*16 + n;
      v2f pv = *(const v2f*)(P + nt*64 + ka);
      float q0 = __builtin_amdgcn_rcpf((float)(nt*64 + ka + 2));
      float q1 = __builtin_amdgcn_rcpf((float)(nt*64 + ka + 3));
      v2f b0, b1;
      b0.x = pv.x * cos2pi(s0f * q0);
      b0.y = pv.y * cos2pi(s0f * q1);
      b1.x = pv.x * cos2pi(s1f * q0);
      b1.y = pv.y * cos2pi(s1f * q1);
      accO0[t] = wmma4(a, b0, accO0[t]);
      accO1[t] = wmma4(a, b1, accO1[t]);
    }
  }

  // ---- writeback: lanes hi=0 carry rows of s0 (accO0), hi=1 rows of s1 (accO1) ----
  if (LAST) {
    #pragma unroll
    for (int t = 0; t < 4; ++t)
      #pragma unroll
      for (int v = 0; v < 8; ++v) {
        float val = hi ? accO1[t][v] : accO0[t][v];
        outg[((size_t)v * S_LEN + (size_t)(s0 + hi)) * D_D + t*16 + n] = val;
      }
  } else {
    #pragma unroll
    for (int t = 0; t < 4; ++t)
      #pragma unroll
      for (int v = 0; v < 8; ++v) {
        float val = hi ? accO1[t][v] : accO0[t][v];
        xb[(v + 8*hi)*XS + t*16 + n] = val;
      }
    lds_fence();
  }
}

__global__ __launch_bounds__(128) void hier_fused_kernel(
    const float* __restrict__ seq,
    const float* __restrict__ M1, const float* __restrict__ P1,
    const float* __restrict__ g1, const float* __restrict__ b1,
    const float* __restrict__ Wr1,
    const float* __restrict__ M2, const float* __restrict__ P2,
    const float* __restrict__ g2, const float* __restrict__ b2,
    const float* __restrict__ M3, const float* __restrict__ P3,
    const float* __restrict__ g3, const float* __restrict__ b3,
    float* __restrict__ out)
{
  __shared__ float lds[NW * WAVE_F];
  const int wave = threadIdx.x >> 5;
  const int lane = threadIdx.x & 31;
  const int s0   = (blockIdx.x * NW + wave) * 2;   // this wave: positions s0, s0+1

  float* wl = lds + wave * WAVE_F;
  float* xb = wl;               // 16 x XS: layer input (rows 0-7: s0, 8-15: s0+1)
  float* tb = wl + 16*XS;       // 16 x TS: pre/post-LN activations
  float* st = tb + 16*TS;       // mu[16], rsig[16]

  // load seq for both positions (16 x 128), coalesced per row
  for (int idx = lane; idx < 16 * IN_D; idx += 32) {
    int m = idx >> 7, k = idx & 127;
    int b = m & 7, s = s0 + (m >> 3);
    xb[m*XS + k] = seq[((size_t)b * S_LEN + (size_t)s) * IN_D + k];
  }
  lds_fence();

  layer_step<IN_D, true,  false>(xb, tb, st, M1, P1, g1, b1, Wr1,    s0, lane, nullptr);
  layer_step<D_D,  false, false>(xb, tb, st, M2, P2, g2, b2, nullptr, s0, lane, nullptr);
  layer_step<D_D,  false, true >(xb, tb, st, M3, P3, g3, b3, nullptr, s0, lane, out);
}

extern "C" void kernel_launch(void* const* d_in, const int* in_sizes, int n_in,
                              void* d_out, int out_size, void* d_ws, size_t ws_size,
                              hipStream_t stream) {
  const float* seq = (const float*)d_in[0];
  const float* M1  = (const float*)d_in[1];
  const float* P1  = (const float*)d_in[2];
  const float* g1  = (const float*)d_in[3];
  const float* b1  = (const float*)d_in[4];
  const float* Wr1 = (const float*)d_in[5];
  const float* M2  = (const float*)d_in[6];
  const float* P2  = (const float*)d_in[7];
  const float* g2  = (const float*)d_in[8];
  const float* b2  = (const float*)d_in[9];
  const float* M3  = (const float*)d_in[10];
  const float* P3  = (const float*)d_in[11];
  const float* g3  = (const float*)d_in[12];
  const float* b3  = (const float*)d_in[13];
  float* out = (float*)d_out;

  dim3 grid(S_LEN / (NW * 2));   // 1024 blocks x 4 waves x 2 positions = 8192
  dim3 block(NW * 32);
  hipLaunchKernelGGL(hier_fused_kernel, grid, block, 0, stream,
                     seq, M1, P1, g1, b1, Wr1, M2, P2, g2, b2, M3, P3, g3, b3, out);
}